// JumpGcnV2_67448166416671
// MI455X (gfx1250) — compile-verified
//
#include <hip/hip_runtime.h>
#include <math.h>

#define NNODES 100000
#define NEDGES 1600000
#define DIN    128
#define HDIM   64
#define NLAYER 4
#define PAIR_STRIDE 160   // dwords per k-pair row of W in LDS.
                          // 160 % 64 == 32 -> the two 16-lane halves (pairs P and
                          // P+1) hit complementary 32-bank sets: conflict-free b64.

typedef __attribute__((ext_vector_type(2))) float v2f;
typedef __attribute__((ext_vector_type(8))) float v8f;

// ---------------------------------------------------------------------------
// Tiled GEMM on the WMMA pipe: each wave computes a 16-row x 64-col tile of
// out = A[rows,K] @ W[K,64] using V_WMMA_F32_16X16X4_F32 (4 N-tiles, shared A
// fragment). W is staged in LDS once per block, k-pair interleaved so each
// B fragment is ONE aligned ds_load_b64 straight into an even VGPR pair
// (element (P=k/2, col, j=k&1) lives at dword P*PAIR_STRIDE + 2*col + j).
// A fragments stream from global as b64. K is a template parameter so the
// k-loop fully unrolls. Epilogue fused per MODE:
//   MODE 0 : out = A@W + bias                                  (proj)
//   MODE 1 : out = relu(LayerNorm(A@W + bias; g,b))            (mlp1/mlp2)
//   MODE 2 : A := 0.9*hi + 0.1*h0 (built on the fly);
//            out = relu(theta*(A@W) + (1-theta)*A); jk = max   (GCNII layer)
// A-fragment layout (16x4 f32): lane m=lane&15 holds row m; VGPR0/1 hold
// K = kbase+{0,1} for lanes 0-15 and K = kbase+{2,3} for lanes 16-31.
// C/D layout: VGPR r holds row r (lanes 0-15, col=lane) / row r+8 (lanes 16-31).
// ---------------------------------------------------------------------------
template <int MODE, int K>
__global__ __launch_bounds__(256) void gemm16_wmma(
    const float* __restrict__ A, const float* __restrict__ W,
    const float* __restrict__ bias, int nrows,
    float* __restrict__ out,
    const float* __restrict__ ln_g, const float* __restrict__ ln_b,
    const float* __restrict__ hi, const float* __restrict__ h0,
    float theta, float* __restrict__ jk, int first_layer)
{
    __shared__ float sW[(K / 2) * PAIR_STRIDE];        // 40 KB max (K=128)

    // Cooperative stage of W[K,64] into LDS, k-pair interleaved.
    for (int i = threadIdx.x; i < K * 32; i += 256) {
        const int r  = i >> 5;                  // source row 0..K-1
        const int c2 = (i & 31) * 2;            // source col 0,2,..,62
        const float2 v = *(const float2*)(W + (size_t)r * HDIM + c2);
        float* d = sW + (r >> 1) * PAIR_STRIDE + (r & 1);
        d[2 * c2]     = v.x;                    // (P, c2,   j=r&1)
        d[2 * c2 + 2] = v.y;                    // (P, c2+1, j=r&1)
    }
    __syncthreads();

    const int lane = threadIdx.x & 31;
    const int wave = threadIdx.x >> 5;
    const int row0 = (blockIdx.x * (blockDim.x >> 5) + wave) * 16;
    if (row0 >= nrows) return;                 // wave-uniform: EXEC stays all-1s
    const int m    = lane & 15;                // row (A) / col (B,C,D) in tile
    const int half = lane >> 4;                // 0: K+0/1, rows 0-7; 1: K+2/3, rows 8-15

    v8f c[4] = {v8f{}, v8f{}, v8f{}, v8f{}};
    const float2* arow2 = (MODE == 2) ? nullptr
                                      : (const float2*)(A + (size_t)(row0 + m) * K);
    const size_t hbase = (size_t)(row0 + m) * HDIM;

#pragma unroll
    for (int k0 = 0; k0 < K; k0 += 4) {
        const int ka = k0 + half * 2;          // even -> 8B-aligned b64 loads
        v2f a;
        if (MODE == 2) {
            const float2 vhi = *(const float2*)(hi + hbase + ka);
            const float2 vh0 = *(const float2*)(h0 + hbase + ka);
            a[0] = 0.9f * vhi.x + 0.1f * vh0.x;
            a[1] = 0.9f * vhi.y + 0.1f * vh0.y;
        } else {
            const float2 va = arow2[ka >> 1];
            a[0] = va.x;
            a[1] = va.y;
        }
        // pair index P = k0/2 + half; fragment = sW[P*STRIDE + 2*col + {0,1}]
        const float* wbase = sW + (size_t)((k0 >> 1) + half) * PAIR_STRIDE + 2 * m;
#pragma unroll
        for (int t = 0; t < 4; ++t) {
            const v2f b = *(const v2f*)(wbase + t * 32);   // one ds_load_b64
            c[t] = __builtin_amdgcn_wmma_f32_16x16x4_f32(
                false, a, false, b, (short)0, c[t], false, false);
        }
    }

    if (MODE == 0) {
#pragma unroll
        for (int t = 0; t < 4; ++t) {
            const int col = t * 16 + m;
            const float bv = bias[col];
#pragma unroll
            for (int r = 0; r < 8; ++r) {
                const int row = row0 + r + half * 8;
                out[(size_t)row * HDIM + col] = c[t][r] + bv;
            }
        }
    } else if (MODE == 1) {
        float y[4][8];
#pragma unroll
        for (int t = 0; t < 4; ++t) {
            const float bv = bias[t * 16 + m];
#pragma unroll
            for (int r = 0; r < 8; ++r) y[t][r] = c[t][r] + bv;
        }
#pragma unroll
        for (int r = 0; r < 8; ++r) {
            float s = y[0][r] + y[1][r] + y[2][r] + y[3][r];
            float q = y[0][r]*y[0][r] + y[1][r]*y[1][r]
                    + y[2][r]*y[2][r] + y[3][r]*y[3][r];
#pragma unroll
            for (int mask = 1; mask < 16; mask <<= 1) {   // stays within 16-lane half
                s += __shfl_xor(s, mask, 32);
                q += __shfl_xor(q, mask, 32);
            }
            const float mean = s * (1.0f / 64.0f);
            const float var  = q * (1.0f / 64.0f) - mean * mean;
            const float rstd = rsqrtf(var + 1e-5f);
            const int   row  = row0 + r + half * 8;
#pragma unroll
            for (int t = 0; t < 4; ++t) {
                const int col = t * 16 + m;
                float v = (y[t][r] - mean) * rstd * ln_g[col] + ln_b[col];
                out[(size_t)row * HDIM + col] = fmaxf(v, 0.0f);
            }
        }
    } else {  // MODE 2: GCNII combine + relu + JumpingKnowledge max
        const float omt = 1.0f - theta;
#pragma unroll
        for (int t = 0; t < 4; ++t) {
            const int col = t * 16 + m;
#pragma unroll
            for (int r = 0; r < 8; ++r) {
                const int row = row0 + r + half * 8;
                const size_t idx = (size_t)row * HDIM + col;
                const float sup = 0.9f * hi[idx] + 0.1f * h0[idx];
                float v = theta * c[t][r] + omt * sup;
                v = fmaxf(v, 0.0f);
                out[idx] = v;
                jk[idx] = first_layer ? v : fmaxf(jk[idx], v);
            }
        }
    }
}

// ---------------------------------------------------------------------------
// SpMM: one wave per edge; lane moves float2 (coalesced 256B/edge gather of
// h[src]) and scatter-adds into hi[dst] with global_atomic_add_f32.
// ---------------------------------------------------------------------------
__global__ __launch_bounds__(256) void spmm_kernel(
    const int* __restrict__ ei, const float* __restrict__ ew,
    const float* __restrict__ h, float* __restrict__ hi, int nedges)
{
    const int gwave = (int)((blockIdx.x * blockDim.x + threadIdx.x) >> 5);
    const int lane  = threadIdx.x & 31;
    if (gwave >= nedges) return;
    const int   src = ei[gwave];
    const int   dst = ei[nedges + gwave];
    const float w   = ew[gwave];
    const float2 v  = ((const float2*)(h + (size_t)src * HDIM))[lane];
    float* o = hi + (size_t)dst * HDIM + lane * 2;
    atomicAdd(o,     v.x * w);
    atomicAdd(o + 1, v.y * w);
}

__global__ __launch_bounds__(256) void zero_kernel(float4* __restrict__ p, int n4)
{
    const int i = blockIdx.x * blockDim.x + threadIdx.x;
    if (i < n4) p[i] = float4{0.f, 0.f, 0.f, 0.f};
}

// ---------------------------------------------------------------------------
// Final: one wave per node; fuse both [64]->[1] heads and the 0.5/0.5 mix.
// ---------------------------------------------------------------------------
__global__ __launch_bounds__(256) void final_kernel(
    const float* __restrict__ jk, const float* __restrict__ m2,
    const float* __restrict__ head_w, const float* __restrict__ head_b,
    const float* __restrict__ w3, const float* __restrict__ b3,
    float* __restrict__ out, int nrows)
{
    const int node = (int)((blockIdx.x * blockDim.x + threadIdx.x) >> 5);
    const int lane = threadIdx.x & 31;
    if (node >= nrows) return;
    const float2 a  = ((const float2*)(jk + (size_t)node * HDIM))[lane];
    const float2 wa = ((const float2*)head_w)[lane];
    const float2 b  = ((const float2*)(m2 + (size_t)node * HDIM))[lane];
    const float2 wb = ((const float2*)w3)[lane];
    float s = a.x * wa.x + a.y * wa.y + b.x * wb.x + b.y * wb.y;
#pragma unroll
    for (int off = 16; off > 0; off >>= 1) s += __shfl_xor(s, off, 32);
    if (lane == 0) out[node] = 0.5f * s + 0.5f * (head_b[0] + b3[0]);
}

extern "C" void kernel_launch(void* const* d_in, const int* in_sizes, int n_in,
                              void* d_out, int out_size, void* d_ws, size_t ws_size,
                              hipStream_t stream)
{
    const float* x        = (const float*)d_in[0];
    const float* ew       = (const float*)d_in[1];
    const float* proj_w   = (const float*)d_in[2];
    const float* proj_b   = (const float*)d_in[3];
    const float* gcn_w    = (const float*)d_in[4];
    const float* mlp_w1   = (const float*)d_in[5];
    const float* mlp_b1   = (const float*)d_in[6];
    const float* ln1_g    = (const float*)d_in[7];
    const float* ln1_b    = (const float*)d_in[8];
    const float* mlp_w2   = (const float*)d_in[9];
    const float* mlp_b2   = (const float*)d_in[10];
    const float* ln2_g    = (const float*)d_in[11];
    const float* ln2_b    = (const float*)d_in[12];
    const float* mlp_w3   = (const float*)d_in[13];
    const float* mlp_b3   = (const float*)d_in[14];
    const float* head_w   = (const float*)d_in[15];
    const float* head_b   = (const float*)d_in[16];
    const int*   eidx     = (const int*)d_in[17];

    const size_t NH = (size_t)NNODES * HDIM;
    float* ws = (float*)d_ws;
    float* hA = ws;            // also holds mlp hidden m (consumed before GCN reuses it)
    float* hB = hA + NH;
    float* h0 = hB + NH;
    float* hi = h0 + NH;
    float* jk = hi + NH;
    float* m2 = jk + NH;

    const dim3 blk(256);
    const int gGemm = (NNODES + 127) / 128;          // 8 waves/block * 16 rows/wave
    const int gZero = (int)((NH / 4 + 255) / 256);
    const int gSpmm = (NEDGES + 7) / 8;              // one wave per edge
    const int gFin  = (NNODES + 7) / 8;

    // MLP branch: m = relu(LN(x@w1+b1)); m2 = relu(LN(m@w2+b2))
    gemm16_wmma<1, DIN><<<gGemm, blk, 0, stream>>>(x, mlp_w1, mlp_b1, NNODES, hA,
                                                   ln1_g, ln1_b, nullptr, nullptr, 0.f, nullptr, 0);
    gemm16_wmma<1, HDIM><<<gGemm, blk, 0, stream>>>(hA, mlp_w2, mlp_b2, NNODES, m2,
                                                    ln2_g, ln2_b, nullptr, nullptr, 0.f, nullptr, 0);
    // GCNII branch: h0 = x@proj_w + proj_b
    gemm16_wmma<0, DIN><<<gGemm, blk, 0, stream>>>(x, proj_w, proj_b, NNODES, h0,
                                                   nullptr, nullptr, nullptr, nullptr, 0.f, nullptr, 0);

    const float* hcur = h0;
    float* hbuf[2] = {hA, hB};
    for (int l = 0; l < NLAYER; ++l) {
        const float theta = logf(1.0f / (float)(l + 1) + 1.0f);  // LAMDA = 1
        zero_kernel<<<gZero, blk, 0, stream>>>((float4*)hi, (int)(NH / 4));
        spmm_kernel<<<gSpmm, blk, 0, stream>>>(eidx, ew, hcur, hi, NEDGES);
        float* hn = hbuf[l & 1];
        gemm16_wmma<2, HDIM><<<gGemm, blk, 0, stream>>>(nullptr, gcn_w + (size_t)l * HDIM * HDIM,
                                                        nullptr, NNODES, hn,
                                                        nullptr, nullptr, hi, h0, theta, jk, l == 0);
        hcur = hn;
    }

    final_kernel<<<gFin, blk, 0, stream>>>(jk, m2, head_w, head_b, mlp_w3, mlp_b3,
                                           (float*)d_out, NNODES);
}